// EmbedMeanField_41970420417062
// MI455X (gfx1250) — compile-verified
//
#include <hip/hip_runtime.h>

typedef float v2f __attribute__((ext_vector_type(2)));
typedef float v8f __attribute__((ext_vector_type(8)));

#define LATENT 128
// K-paired LDS weight layout: lwp[kpair*KP_STRIDE + col*2 + {0,1}] = W[2kp+{0,1}][col]
// KP_STRIDE floats per kpair; 288 % 64 == 32 -> half-waves (kp vs kp+1) use disjoint bank ranges.
#define KP_STRIDE 288

__device__ __forceinline__ void atomAddF(float* p, float v) {
    unsafeAtomicAdd(p, v);   // hardware global_atomic_add_f32 on CDNA5
}

// ---------------------------------------------------------------------------
// Dense GEMM: C[nrows,128] = op( A[nrows,128] @ W[128,128] + bias (+ skip) )
// One 256-thread block = 8 waves = 128 rows; each wave owns a 16x128 strip.
// f32 WMMA 16x16x4; B-fragments are single ds_load_b64 from K-paired LDS.
// ---------------------------------------------------------------------------
template <bool RELU, bool SKIP>
__global__ __launch_bounds__(256) void gemm128_kernel(
    const float* __restrict__ A, const float* __restrict__ W,
    const float* __restrict__ bias, const float* __restrict__ skip,
    float* __restrict__ C, int nrows)
{
    __shared__ float lwp[64 * KP_STRIDE];   // 64 kpairs (K=128)
    __shared__ float lb[LATENT];

    const int tid = threadIdx.x;
    // cooperative load of W into K-paired layout; each i is one (kpair, col) b64
    for (int i = tid; i < 64 * 128; i += 256) {
        int kp = i >> 7;        // kpair
        int c  = i & 127;       // col
        v2f w;
        w.x = W[(2 * kp)     * 128 + c];
        w.y = W[(2 * kp + 1) * 128 + c];
        *(v2f*)&lwp[kp * KP_STRIDE + c * 2] = w;
    }
    if (tid < LATENT) lb[tid] = bias[tid];
    __syncthreads();

    const int wave  = tid >> 5;
    const int lane  = tid & 31;
    const int laneM = lane & 15;
    const int kSel  = (lane >> 4) << 1;  // lanes 0-15: K=k0,k0+1 ; lanes 16-31: K=k0+2,k0+3

    const long rowBase = (long)blockIdx.x * 128 + wave * 16;
    long row  = rowBase + laneM;
    long rowC = row < nrows ? row : (long)nrows - 1;  // clamp: garbage rows are discarded
    const float* aRow = A + rowC * LATENT;

    v8f acc[8] = {};

    for (int k0 = 0; k0 < 128; k0 += 4) {
        v2f a = *(const v2f*)&aRow[k0 + kSel];
        const float* bBase = &lwp[((k0 + kSel) >> 1) * KP_STRIDE + laneM * 2];
#pragma unroll
        for (int t = 0; t < 8; ++t) {
            v2f b = *(const v2f*)&bBase[t * 32];   // ds_load_b64, even-aligned pair
            acc[t] = __builtin_amdgcn_wmma_f32_16x16x4_f32(
                false, a, false, b, (short)0, acc[t], false, false);
        }
    }

    // epilogue: C/D layout -> lane<16: VGPR r is (M=r, N=lane); lane>=16: (M=r+8, N=lane-16)
    const int rOff = (lane < 16) ? 0 : 8;
#pragma unroll
    for (int t = 0; t < 8; ++t) {
        int col = t * 16 + laneM;
#pragma unroll
        for (int r = 0; r < 8; ++r) {
            long m = rowBase + rOff + r;
            if (m < nrows) {
                float v = acc[t][r] + lb[col];
                if (SKIP) v += skip[m * LATENT + col];
                if (RELU) v = v > 0.f ? v : 0.f;
                C[m * LATENT + col] = v;
            }
        }
    }
}

// ---------------------------------------------------------------------------
// Edge embedding + scatter: edge_lin = edge_feat[E,16] @ w_e2l[16,128] + b,
// then IM[edge_dst[e]] += edge_lin[e]  (f32 atomics).
// One wave = 16 edges; K-loop of 4 WMMA steps.
// ---------------------------------------------------------------------------
__global__ __launch_bounds__(256) void edge_embed_kernel(
    const float* __restrict__ EF, const float* __restrict__ W,
    const float* __restrict__ bias, const int* __restrict__ dst,
    float* __restrict__ IM, int E)
{
    __shared__ float lwp[8 * KP_STRIDE];    // 8 kpairs (K=16)
    __shared__ float lb[LATENT];

    const int tid = threadIdx.x;
    for (int i = tid; i < 8 * 128; i += 256) {
        int kp = i >> 7;
        int c  = i & 127;
        v2f w;
        w.x = W[(2 * kp)     * 128 + c];
        w.y = W[(2 * kp + 1) * 128 + c];
        *(v2f*)&lwp[kp * KP_STRIDE + c * 2] = w;
    }
    if (tid < LATENT) lb[tid] = bias[tid];
    __syncthreads();

    const int wave  = tid >> 5;
    const int lane  = tid & 31;
    const int laneM = lane & 15;
    const int kSel  = (lane >> 4) << 1;

    const long eBase = (long)blockIdx.x * 128 + wave * 16;
    long e  = eBase + laneM;
    long eC = e < E ? e : (long)E - 1;
    const float* aRow = EF + eC * 16;

    v8f acc[8] = {};

#pragma unroll
    for (int k0 = 0; k0 < 16; k0 += 4) {
        v2f a = *(const v2f*)&aRow[k0 + kSel];
        const float* bBase = &lwp[((k0 + kSel) >> 1) * KP_STRIDE + laneM * 2];
#pragma unroll
        for (int t = 0; t < 8; ++t) {
            v2f b = *(const v2f*)&bBase[t * 32];
            acc[t] = __builtin_amdgcn_wmma_f32_16x16x4_f32(
                false, a, false, b, (short)0, acc[t], false, false);
        }
    }

    const int rOff = (lane < 16) ? 0 : 8;
    int drow[8];
#pragma unroll
    for (int r = 0; r < 8; ++r) {
        long m = eBase + rOff + r;
        drow[r] = (m < E) ? dst[m] : -1;
    }
#pragma unroll
    for (int t = 0; t < 8; ++t) {
        int col = t * 16 + laneM;
#pragma unroll
        for (int r = 0; r < 8; ++r) {
            if (drow[r] >= 0) {
                atomAddF(&IM[(long)drow[r] * LATENT + col], acc[t][r] + lb[col]);
            }
        }
    }
}

// ---------------------------------------------------------------------------
// n2n SpMM: pool[dst[e]] += h[src[e]]   (wave per edge, float4 per lane)
// ---------------------------------------------------------------------------
__global__ __launch_bounds__(256) void spmm_gather_kernel(
    const float* __restrict__ h, const int* __restrict__ src,
    const int* __restrict__ dst, float* __restrict__ pool, int E)
{
    long e = (long)blockIdx.x * 8 + (threadIdx.x >> 5);
    if (e >= E) return;
    int lane = threadIdx.x & 31;
    int s = src[e], d = dst[e];
    float4 v = ((const float4*)(h + (long)s * LATENT))[lane];
    float* out = pool + (long)d * LATENT + lane * 4;
    atomAddF(out + 0, v.x);
    atomAddF(out + 1, v.y);
    atomAddF(out + 2, v.z);
    atomAddF(out + 3, v.w);
}

// ---------------------------------------------------------------------------
// Subgraph pooling: y[graph_ids[n]] += act[n]   (wave per node)
// ---------------------------------------------------------------------------
__global__ __launch_bounds__(256) void graph_pool_kernel(
    const float* __restrict__ act, const int* __restrict__ gid,
    float* __restrict__ y, int N)
{
    long n = (long)blockIdx.x * 8 + (threadIdx.x >> 5);
    if (n >= N) return;
    int lane = threadIdx.x & 31;
    int g = gid[n];
    float4 v = ((const float4*)(act + n * LATENT))[lane];
    float* out = y + (long)g * LATENT + lane * 4;
    atomAddF(out + 0, v.x);
    atomAddF(out + 1, v.y);
    atomAddF(out + 2, v.z);
    atomAddF(out + 3, v.w);
}

__global__ void relu_copy4_kernel(const float4* __restrict__ in,
                                  float4* __restrict__ out, long n4)
{
    long i = (long)blockIdx.x * blockDim.x + threadIdx.x;
    if (i < n4) {
        float4 v = in[i];
        v.x = fmaxf(v.x, 0.f); v.y = fmaxf(v.y, 0.f);
        v.z = fmaxf(v.z, 0.f); v.w = fmaxf(v.w, 0.f);
        out[i] = v;
    }
}

__global__ void relu_inplace_kernel(float* y, int n)
{
    int i = blockIdx.x * blockDim.x + threadIdx.x;
    if (i < n) { float v = y[i]; y[i] = v > 0.f ? v : 0.f; }
}

// ---------------------------------------------------------------------------
extern "C" void kernel_launch(void* const* d_in, const int* in_sizes, int n_in,
                              void* d_out, int out_size, void* d_ws, size_t ws_size,
                              hipStream_t stream)
{
    const float* node_feat = (const float*)d_in[0];
    const float* edge_feat = (const float*)d_in[1];
    const int*   edge_src  = (const int*)d_in[2];
    const int*   edge_dst  = (const int*)d_in[3];
    const int*   graph_ids = (const int*)d_in[4];
    const float* w_n2l     = (const float*)d_in[5];
    const float* b_n2l     = (const float*)d_in[6];
    const float* w_e2l     = (const float*)d_in[7];
    const float* b_e2l     = (const float*)d_in[8];
    const float* w_conv    = (const float*)d_in[9];
    const float* b_conv    = (const float*)d_in[10];
    const float* w_out     = (const float*)d_in[11];
    const float* b_out     = (const float*)d_in[12];

    const int N = in_sizes[0] / LATENT;   // 100000
    const int E = in_sizes[1] / 16;       // 1600000

    float* IM   = (float*)d_ws;                 // input_message (skip connection)
    float* h    = IM + (size_t)N * LATENT;      // node state
    float* pool = h  + (size_t)N * LATENT;      // n2npool / reluact scratch
    float* y    = (float*)d_out;

    dim3 blk(256);
    const int gemmGrid = (N + 127) / 128;
    const int edgeGrid = (E + 127) / 128;
    const long nElem   = (long)N * LATENT;

    // 1) IM = node_feat @ w_n2l + b_n2l
    gemm128_kernel<false, false><<<gemmGrid, blk, 0, stream>>>(
        node_feat, w_n2l, b_n2l, nullptr, IM, N);

    // 2) IM += scatter(edge_feat @ w_e2l + b_e2l, edge_dst)
    edge_embed_kernel<<<edgeGrid, blk, 0, stream>>>(
        edge_feat, w_e2l, b_e2l, edge_dst, IM, E);

    // 3) h = relu(IM)
    relu_copy4_kernel<<<(int)((nElem / 4 + 255) / 256), blk, 0, stream>>>(
        (const float4*)IM, (float4*)h, nElem / 4);

    // 4) mean-field iterations
    for (int lv = 0; lv < 3; ++lv) {
        hipMemsetAsync(pool, 0, (size_t)N * LATENT * sizeof(float), stream);
        spmm_gather_kernel<<<(E + 7) / 8, blk, 0, stream>>>(
            h, edge_src, edge_dst, pool, E);
        // h = relu(pool @ w_conv + b_conv + IM)
        gemm128_kernel<true, true><<<gemmGrid, blk, 0, stream>>>(
            pool, w_conv, b_conv, IM, h, N);
    }

    // 5) reluact = relu(h @ w_out + b_out)  -> pool
    gemm128_kernel<true, false><<<gemmGrid, blk, 0, stream>>>(
        h, w_out, b_out, nullptr, pool, N);

    // 6) y = relu(segment_sum(reluact, graph_ids))
    hipMemsetAsync(d_out, 0, (size_t)out_size * sizeof(float), stream);
    graph_pool_kernel<<<(N + 7) / 8, blk, 0, stream>>>(pool, graph_ids, y, N);
    relu_inplace_kernel<<<(out_size + 255) / 256, blk, 0, stream>>>(y, out_size);
}